// STAGGCNModel_1082331758980
// MI455X (gfx1250) — compile-verified
//
#include <hip/hip_runtime.h>
#include <math.h>

#define IN_CH 128
#define OUT_CH 32
#define HEADS 3
#define HC (HEADS * OUT_CH)   // 96
#define SLOPE 0.2f

typedef __attribute__((ext_vector_type(2))) float v2f;
typedef __attribute__((ext_vector_type(8))) float v8f;

// ---------------------------------------------------------------------------
// Monotone float <-> uint ordering (for atomicMax on signed floats)
// ---------------------------------------------------------------------------
__device__ __forceinline__ unsigned f2ord(float x) {
  unsigned u = __float_as_uint(x);
  return (u & 0x80000000u) ? ~u : (u | 0x80000000u);
}
__device__ __forceinline__ float ord2f(unsigned k) {
  unsigned u = (k & 0x80000000u) ? (k & 0x7fffffffu) : ~k;
  return __uint_as_float(u);
}

// ---------------------------------------------------------------------------
// 1) h[N,96] = x[N,128] @ W^T   (W is [96,128])  via V_WMMA_F32_16X16X4_F32
//    One wave32 computes one 16x16 output tile; K loop of 32 WMMA ops.
//    A 16x4 f32: lane m = lane&15, VGPR0/1 hold K = 2*(lane>>4) + {0,1}.
//    B 4x16 f32: lane n = lane&15, same K striping.
//    D 16x16 f32: VGPR r -> row r (lanes 0-15) / r+8 (lanes 16-31), col lane&15.
// ---------------------------------------------------------------------------
__global__ __launch_bounds__(256) void gat_gemm_wmma(
    const float* __restrict__ x, const float* __restrict__ W,
    float* __restrict__ h, int N, int nMTiles) {
  int wave = (blockIdx.x * blockDim.x + threadIdx.x) >> 5;
  int lane = threadIdx.x & 31;
  int nTiles = nMTiles * (HC / 16);
  if (wave >= nTiles) return;
  int mt = wave / (HC / 16);
  int nt = wave % (HC / 16);
  int mBase = mt * 16, nBase = nt * 16;

  int rc = lane & 15;            // A row / B col within tile
  int kb = (lane >> 4) << 1;     // K sub-offset: 0 or 2
  int mrow = mBase + rc; if (mrow >= N) mrow = N - 1;   // clamp (N%16==0 anyway)
  const float* __restrict__ xrow = x + (size_t)mrow * IN_CH;
  const float* __restrict__ wrow = W + (size_t)(nBase + rc) * IN_CH;

  v8f c = {0.f, 0.f, 0.f, 0.f, 0.f, 0.f, 0.f, 0.f};
#pragma unroll
  for (int k0 = 0; k0 < IN_CH; k0 += 4) {
    v2f a, b;
    a.x = xrow[k0 + kb]; a.y = xrow[k0 + kb + 1];
    b.x = wrow[k0 + kb]; b.y = wrow[k0 + kb + 1];
    // (neg_a, A, neg_b, B, c_mod, C, reuse_a, reuse_b)
    c = __builtin_amdgcn_wmma_f32_16x16x4_f32(false, a, false, b, (short)0, c,
                                              false, false);
  }
  int col = lane & 15;
  int radd = (lane >> 4) << 3;   // 0 or 8
#pragma unroll
  for (int r = 0; r < 8; ++r) {
    int row = mBase + r + radd;
    if (row < N) h[(size_t)row * HC + nBase + col] = c[r];
  }
}

// ---------------------------------------------------------------------------
// 2) per-node attention scores: sj[n,h] = h[n,h,:].attn_j ; si likewise
// ---------------------------------------------------------------------------
__global__ void gat_scores(const float* __restrict__ h,
                           const float* __restrict__ aj,
                           const float* __restrict__ ai,
                           float* __restrict__ sj, float* __restrict__ si,
                           int N) {
  int t = blockIdx.x * blockDim.x + threadIdx.x;  // t = n*HEADS + head
  if (t >= N * HEADS) return;
  int n = t / HEADS, hd = t % HEADS;
  const float* hr = h + (size_t)n * HC + hd * OUT_CH;
  const float* ajr = aj + hd * OUT_CH;
  const float* air = ai + hd * OUT_CH;
  float accj = 0.f, acci = 0.f;
#pragma unroll
  for (int cch = 0; cch < OUT_CH; ++cch) {
    float v = hr[cch];
    accj = fmaf(v, ajr[cch], accj);
    acci = fmaf(v, air[cch], acci);
  }
  sj[t] = accj;
  si[t] = acci;
}

// ---------------------------------------------------------------------------
// 3) init: segmax keys = 0 (== ordered -max), segsum = 0, out = 0
// ---------------------------------------------------------------------------
__global__ void gat_init(unsigned* __restrict__ segmax,
                         float* __restrict__ segsum, float* __restrict__ out,
                         int N) {
  int t = blockIdx.x * blockDim.x + threadIdx.x;
  if (t < N * HEADS) { segmax[t] = 0u; segsum[t] = 0.f; }
  if (t < N * OUT_CH) out[t] = 0.f;
}

// ---------------------------------------------------------------------------
// 4) per-edge: alpha = leaky_relu(sj[src[ej[e]]] + si[dst[ei[e]]]); seg max
// ---------------------------------------------------------------------------
__global__ void gat_alpha_max(const int* __restrict__ src,
                              const int* __restrict__ dst,
                              const float* __restrict__ sj,
                              const float* __restrict__ si,
                              float* __restrict__ alpha,
                              unsigned* __restrict__ segmax, int E, int Et) {
  int e = blockIdx.x * blockDim.x + threadIdx.x;
  if (e >= Et) return;
  int je = (e < E) ? src[e] : (e - E);
  int ie = (e < E) ? dst[e] : (e - E);
  int jj = src[je];   // ej[ej[e]] : je in [0,N) < E -> first-E region = src
  int ii = dst[ie];   // ei[ei[e]]
#pragma unroll
  for (int hd = 0; hd < HEADS; ++hd) {
    float a = sj[jj * HEADS + hd] + si[ii * HEADS + hd];
    a = (a >= 0.f) ? a : SLOPE * a;           // leaky_relu
    alpha[(size_t)e * HEADS + hd] = a;
    atomicMax(&segmax[ie * HEADS + hd], f2ord(a));
  }
}

// ---------------------------------------------------------------------------
// 5) per-edge: ex = exp(alpha - segmax[ie]); segsum += ex (alpha overwritten)
// ---------------------------------------------------------------------------
__global__ void gat_exp_sum(const int* __restrict__ dst,
                            float* __restrict__ alpha,
                            const unsigned* __restrict__ segmax,
                            float* __restrict__ segsum, int E, int Et) {
  int e = blockIdx.x * blockDim.x + threadIdx.x;
  if (e >= Et) return;
  int ie = (e < E) ? dst[e] : (e - E);
#pragma unroll
  for (int hd = 0; hd < HEADS; ++hd) {
    float mx = ord2f(segmax[ie * HEADS + hd]);
    float ex = expf(alpha[(size_t)e * HEADS + hd] - mx);
    alpha[(size_t)e * HEADS + hd] = ex;
    atomicAdd(&segsum[ie * HEADS + hd], ex);
  }
}

// ---------------------------------------------------------------------------
// 6) scatter: out[ie,:] += h[je,hd,:] * (ex/den) / HEADS  (head mean folded in)
// ---------------------------------------------------------------------------
__global__ void gat_scatter(const int* __restrict__ src,
                            const int* __restrict__ dst,
                            const float* __restrict__ h,
                            const float* __restrict__ alpha,
                            const float* __restrict__ segsum,
                            float* __restrict__ out, int E, int Et) {
  int t = blockIdx.x * blockDim.x + threadIdx.x;  // t = e*HEADS + head
  if (t >= Et * HEADS) return;
  int e = t / HEADS, hd = t % HEADS;
  int je = (e < E) ? src[e] : (e - E);
  int ie = (e < E) ? dst[e] : (e - E);
  float w = alpha[t] / segsum[ie * HEADS + hd] * (1.0f / HEADS);
  const float4* hr = (const float4*)(h + (size_t)je * HC + hd * OUT_CH);
  float* orow = out + (size_t)ie * OUT_CH;
#pragma unroll
  for (int q = 0; q < OUT_CH / 4; ++q) {
    float4 v = hr[q];
    atomicAdd(&orow[q * 4 + 0], v.x * w);
    atomicAdd(&orow[q * 4 + 1], v.y * w);
    atomicAdd(&orow[q * 4 + 2], v.z * w);
    atomicAdd(&orow[q * 4 + 3], v.w * w);
  }
}

// ---------------------------------------------------------------------------
// 7) out += bias
// ---------------------------------------------------------------------------
__global__ void gat_bias(float* __restrict__ out, const float* __restrict__ bias,
                         int N) {
  int t = blockIdx.x * blockDim.x + threadIdx.x;
  if (t >= N * OUT_CH) return;
  out[t] += bias[t & (OUT_CH - 1)];
}

// ---------------------------------------------------------------------------
extern "C" void kernel_launch(void* const* d_in, const int* in_sizes, int n_in,
                              void* d_out, int out_size, void* d_ws,
                              size_t ws_size, hipStream_t stream) {
  const float* x      = (const float*)d_in[0];
  const int*   eidx   = (const int*)d_in[1];
  const float* W      = (const float*)d_in[2];
  const float* attn_j = (const float*)d_in[3];
  const float* attn_i = (const float*)d_in[4];
  const float* bias   = (const float*)d_in[5];

  int N  = in_sizes[0] / IN_CH;
  int E  = in_sizes[1] / 2;
  int Et = E + N;
  const int* src = eidx;
  const int* dst = eidx + E;

  // workspace layout (floats): h | sj | si | alpha | segmax | segsum
  float* ws     = (float*)d_ws;
  float* h      = ws;
  float* sj     = h + (size_t)N * HC;
  float* si     = sj + (size_t)N * HEADS;
  float* alpha  = si + (size_t)N * HEADS;
  unsigned* segmax = (unsigned*)(alpha + (size_t)Et * HEADS);
  float* segsum = (float*)(segmax + (size_t)N * HEADS);
  float* out    = (float*)d_out;

  // 1) GEMM via fp32 WMMA
  int nMTiles = (N + 15) / 16;
  int tiles = nMTiles * (HC / 16);
  int wavesPerBlock = 8;  // 256 threads
  gat_gemm_wmma<<<(tiles + wavesPerBlock - 1) / wavesPerBlock,
                  wavesPerBlock * 32, 0, stream>>>(x, W, h, N, nMTiles);

  // 2) per-node scores
  int ts = N * HEADS;
  gat_scores<<<(ts + 255) / 256, 256, 0, stream>>>(h, attn_j, attn_i, sj, si, N);

  // 3) init stats + output
  int ti = N * OUT_CH;
  gat_init<<<(ti + 255) / 256, 256, 0, stream>>>(segmax, segsum, out, N);

  // 4) leaky-relu scores + segment max
  gat_alpha_max<<<(Et + 255) / 256, 256, 0, stream>>>(src, dst, sj, si, alpha,
                                                      segmax, E, Et);
  // 5) exp + segment sum
  gat_exp_sum<<<(Et + 255) / 256, 256, 0, stream>>>(dst, alpha, segmax, segsum,
                                                    E, Et);
  // 6) weighted scatter (head mean folded in)
  int tsc = Et * HEADS;
  gat_scatter<<<(tsc + 255) / 256, 256, 0, stream>>>(src, dst, h, alpha,
                                                     segsum, out, E, Et);
  // 7) bias
  gat_bias<<<(ti + 255) / 256, 256, 0, stream>>>(out, bias, N);
}